// SGC_74045236183109
// MI455X (gfx1250) — compile-verified
//
#include <hip/hip_runtime.h>
#include <hip/hip_bf16.h>

typedef __attribute__((ext_vector_type(2))) float v2f;
typedef __attribute__((ext_vector_type(8))) float v8f;

// ---------------------------------------------------------------------------
// g0[N,64] = x[N,128] @ W[64,128]^T   via V_WMMA_F32_16X16X4_F32
// grid.x = N/16 (m-tiles), block = 128 threads = 4 waves (one per n-tile).
// ---------------------------------------------------------------------------
__global__ void sgc_gemm_xWT(const float* __restrict__ x,
                             const float* __restrict__ W,
                             float* __restrict__ g0) {
    const int lane   = threadIdx.x & 31;
    const int n_tile = threadIdx.x >> 5;     // 0..3
    const int m_tile = blockIdx.x;           // 0..N/16-1
    const int lane16 = lane & 15;
    const int hi     = lane >> 4;            // 0 or 1

    // A frag source row (x) and B frag source row (W); k-pair offset by 2*hi.
    const float* __restrict__ xrow = x + (size_t)(m_tile * 16 + lane16) * 128 + hi * 2;
    const float* __restrict__ wrow = W + (size_t)(n_tile * 16 + lane16) * 128 + hi * 2;

    v8f c = {};
#pragma unroll
    for (int k = 0; k < 128; k += 4) {
        v2f a, b;
        a.x = xrow[k];
        a.y = xrow[k + 1];
        b.x = wrow[k];
        b.y = wrow[k + 1];
        // (neg_a, A, neg_b, B, c_mod, C, reuse_a, reuse_b)
        c = __builtin_amdgcn_wmma_f32_16x16x4_f32(false, a, false, b,
                                                  (short)0, c, false, false);
    }

    // C/D layout: vgpr r -> m = 16*m_tile + r + 8*hi, n = 16*n_tile + lane16
    float* __restrict__ dst = g0 + (size_t)(m_tile * 16 + hi * 8) * 64
                                 + n_tile * 16 + lane16;
#pragma unroll
    for (int r = 0; r < 8; ++r) {
        dst[(size_t)r * 64] = c[r];
    }
}

// ---------------------------------------------------------------------------
// One COO SpMM hop on 64-wide features: gout[row[e]] += vals[e] * gin[col[e]]
// One wave per edge; each lane owns 2 consecutive features (float2 gather,
// two global_atomic_add_f32 scatters).
// ---------------------------------------------------------------------------
__global__ void sgc_spmm_scatter(const int* __restrict__ row,
                                 const int* __restrict__ col,
                                 const float* __restrict__ vals,
                                 const float* __restrict__ gin,
                                 float* __restrict__ gout,
                                 int E) {
    const int gid = blockIdx.x * blockDim.x + threadIdx.x;
    const int e   = gid >> 5;
    if (e >= E) return;                       // wave-uniform guard
    const int lane = gid & 31;

    const int   c = col[e];
    const int   r = row[e];
    const float v = vals[e];

    const float2 s = *((const float2*)(gin + (size_t)c * 64) + lane);
    float* dst = gout + (size_t)r * 64 + lane * 2;
    atomicAdd(dst,     v * s.x);
    atomicAdd(dst + 1, v * s.y);
}

// ---------------------------------------------------------------------------
// Accumulator initializers (d_ws / d_out are poisoned by the harness).
// ---------------------------------------------------------------------------
__global__ void sgc_zero(float* __restrict__ p, int n) {
    int i = blockIdx.x * blockDim.x + threadIdx.x;
    if (i < n) p[i] = 0.0f;
}

__global__ void sgc_init_bias(const float* __restrict__ b,
                              float* __restrict__ out, int n) {
    int i = blockIdx.x * blockDim.x + threadIdx.x;
    if (i < n) out[i] = b[i & 63];
}

// ---------------------------------------------------------------------------
// out = A @ (A @ (x @ W^T)) + b      (linear-op reassociation of reference)
// ---------------------------------------------------------------------------
extern "C" void kernel_launch(void* const* d_in, const int* in_sizes, int n_in,
                              void* d_out, int out_size, void* d_ws, size_t ws_size,
                              hipStream_t stream) {
    const float* x    = (const float*)d_in[0];
    const int*   row  = (const int*)  d_in[1];
    const int*   col  = (const int*)  d_in[2];
    const float* vals = (const float*)d_in[3];
    const float* W    = (const float*)d_in[4];
    const float* b    = (const float*)d_in[5];
    float*       out  = (float*)d_out;

    const int N = in_sizes[0] / 128;   // 50000, divisible by 16
    const int E = in_sizes[1];         // 600000

    float* g0 = (float*)d_ws;                  // [N,64]
    float* g1 = g0 + (size_t)N * 64;           // [N,64]

    const int nf   = N * 64;
    const int blkF = (nf + 255) / 256;
    const int blkE = (E * 32 + 255) / 256;

    // 1) g0 = x @ W^T  (WMMA f32)
    sgc_gemm_xWT<<<N / 16, 128, 0, stream>>>(x, W, g0);

    // 2) g1 = A @ g0
    sgc_zero<<<blkF, 256, 0, stream>>>(g1, nf);
    sgc_spmm_scatter<<<blkE, 256, 0, stream>>>(row, col, vals, g0, g1, E);

    // 3) out = A @ g1 + b
    sgc_init_bias<<<blkF, 256, 0, stream>>>(b, out, nf);
    sgc_spmm_scatter<<<blkE, 256, 0, stream>>>(row, col, vals, g1, out, E);
}